// AttentionNet_88210038325548
// MI455X (gfx1250) — compile-verified
//
#include <hip/hip_runtime.h>
#include <hip/hip_bf16.h>
#include <math.h>

// ---------------------------------------------------------------------------
// AttentionNet cross-modal attention block for MI455X (gfx1250, wave32).
//
// Key algebraic reduction: both attention outputs are only mean-pooled over
// the spatial axis, so we only ever need the COLUMN SUMS of the softmax
// matrix: s[b,n] = sum_m att[b,m,n]. The [B,4096,4096] att tensor is never
// materialized; the vp@att^T / vi@att^T GEMMs collapse to mat-vecs.
// Roofline: ~1.5 GB HBM traffic (pc2d streamed twice) -> ~63us @ 23.3 TB/s;
// ~215 GF of fp32 WMMA (V_WMMA_F32_16X16X4_F32) -> compute is negligible.
// Memory bound => keep full fp32 fidelity.
// ---------------------------------------------------------------------------

typedef float v2f __attribute__((ext_vector_type(2)));
typedef float v8f __attribute__((ext_vector_type(8)));

#define BB        16
#define CQ_DIM    256
#define CK_DIM    2048
#define NSP       4096     // H*W
#define FUSED_DIM 2304
#define HID       1024
#define NCLS      40

__device__ __forceinline__ float wave_reduce_sum(float v) {
#pragma unroll
  for (int off = 16; off > 0; off >>= 1) v += __shfl_down(v, off, 32);
  return v;
}

__global__ void zero_kernel(float* __restrict__ p, int n) {
  int i = blockIdx.x * blockDim.x + threadIdx.x;
  if (i < n) p[i] = 0.0f;
}

// ---------------------------------------------------------------------------
// conv1x1 as GEMM with V_WMMA_F32_16X16X4_F32.
// D[b,m,n] = sum_k W[m,k] * X[b,k,n] + bias[m]
//   A frag (16x4): lanes 0-15 -> M=lane, v0/v1 = K 0/1; lanes 16-31 -> K 2/3.
//   B frag (4x16): lanes 0-15 -> N=lane, v0/v1 = K 0/1; lanes 16-31 -> K 2/3.
//   D (8 VGPRs)  : vgpr i = row m0 + 8*half + i, col = n0 + (lane&15).
// Block = 4 waves, each wave owns a 16(m) x 64(n) strip (A reused over 4
// n-tiles). transpose_out=1 stores D^T (used for q -> [B, N, CQ] layout).
// ---------------------------------------------------------------------------
__global__ __launch_bounds__(128)
void conv1x1_wmma_kernel(const float* __restrict__ W,
                         const float* __restrict__ X,
                         const float* __restrict__ bias,
                         float* __restrict__ out,
                         int M, int K, int transpose_out)
{
  const int lane = threadIdx.x & 31;
  const int wave = threadIdx.x >> 5;
  const int half = lane >> 4;
  const int l16  = lane & 15;
  const int b    = blockIdx.z;
  const int m0   = blockIdx.y * 64 + wave * 16;
  const int n0   = blockIdx.x * 64;

  const float* __restrict__ Wrow = W + (size_t)(m0 + l16) * K;
  const float* __restrict__ Xb   = X + (size_t)b * K * NSP;

  v8f zc = {};
  v8f acc[4];
#pragma unroll
  for (int t = 0; t < 4; ++t) acc[t] = zc;

  for (int k0 = 0; k0 < K; k0 += 4) {
    v2f a = *(const v2f*)(Wrow + k0 + 2 * half);
    const float* __restrict__ Xp =
        Xb + (size_t)(k0 + 2 * half) * NSP + n0 + l16;
    // stream-ahead prefetch of the X (pc2d/img) tile, 32 K-rows ahead
    if (k0 + 32 < K)
      __builtin_prefetch(Xp + (size_t)32 * NSP, 0, 0);
#pragma unroll
    for (int t = 0; t < 4; ++t) {
      v2f bf;
      bf.x = Xp[t * 16];
      bf.y = Xp[t * 16 + NSP];
      acc[t] = __builtin_amdgcn_wmma_f32_16x16x4_f32(
          false, a, false, bf, (short)0, acc[t], false, false);
    }
  }

#pragma unroll
  for (int t = 0; t < 4; ++t) {
#pragma unroll
    for (int i = 0; i < 8; ++i) {
      const int row = m0 + half * 8 + i;
      const int col = n0 + t * 16 + l16;
      const float v = acc[t][i] + bias[row];
      if (!transpose_out)
        out[(size_t)b * M * NSP + (size_t)row * NSP + col] = v;
      else
        out[(size_t)b * NSP * M + (size_t)col * M + row] = v;
    }
  }
}

// ---------------------------------------------------------------------------
// Per (batch, 16-row block): logits = q[16,256] @ k[256,4096] via WMMA into
// 256 KB LDS (<= 320 KB WGP budget), row softmax in LDS, then fold column
// sums into s[b, 4096] with global atomics. att never hits HBM.
// 8 waves; wave w computes the 512-column strip [w*512, w*512+512).
// All 64 A fragments (the block's 16 q-rows x 256 K) are hoisted into
// registers once (128 VGPRs) -- this kernel runs 1 block/WGP due to LDS,
// so VGPR pressure is free and the inner loop is pure B-load + WMMA.
// ---------------------------------------------------------------------------
__global__ __launch_bounds__(256)
void attn_colsum_kernel(const float* __restrict__ q,   // [B, N, CQ]
                        const float* __restrict__ kk,  // [B, CQ, N]
                        float* __restrict__ s)         // [B, N]
{
  extern __shared__ float smem[];       // 16 * 4096 floats
  __shared__ float red[16 * 16];
  __shared__ float rowmax[16];
  __shared__ float rowinv[16];

  const int tid  = threadIdx.x;
  const int lane = tid & 31;
  const int wave = tid >> 5;
  const int half = lane >> 4;
  const int l16  = lane & 15;
  const int b    = blockIdx.y;
  const int m0   = blockIdx.x * 16;

  const float* __restrict__ Arow =
      q + (size_t)b * NSP * CQ_DIM + (size_t)(m0 + l16) * CQ_DIM;
  const float* __restrict__ kb = kk + (size_t)b * CQ_DIM * NSP;

  // hoist all A fragments for this block's 16 rows (K = 256 -> 64 frags)
  v2f a[64];
#pragma unroll
  for (int k = 0; k < 64; ++k)
    a[k] = *(const v2f*)(Arow + 4 * k + 2 * half);

  v8f zc = {};
  for (int nt = 0; nt < 32; ++nt) {
    const int n0 = wave * 512 + nt * 16;
    const float* __restrict__ Bbase = kb + n0 + l16 + (size_t)(2 * half) * NSP;
    v8f acc = zc;
#pragma unroll
    for (int k = 0; k < 64; ++k) {
      v2f bf;
      bf.x = Bbase[(size_t)(4 * k) * NSP];
      bf.y = Bbase[(size_t)(4 * k + 1) * NSP];
      acc = __builtin_amdgcn_wmma_f32_16x16x4_f32(
          false, a[k], false, bf, (short)0, acc, false, false);
    }
#pragma unroll
    for (int i = 0; i < 8; ++i)
      smem[(half * 8 + i) * NSP + n0 + l16] = acc[i];
  }
  __syncthreads();

  // row max (16 rows x 16 segments of 256)
  {
    const int row = tid & 15, seg = tid >> 4;
    const float* p = smem + row * NSP + seg * 256;
    float m = -3.4e38f;
    for (int j = 0; j < 256; ++j) m = fmaxf(m, p[j]);
    red[row * 16 + seg] = m;
  }
  __syncthreads();
  if (tid < 16) {
    float m = -3.4e38f;
    for (int sg = 0; sg < 16; ++sg) m = fmaxf(m, red[tid * 16 + sg]);
    rowmax[tid] = m;
  }
  __syncthreads();

  // exp in place + row sums
  {
    const int row = tid & 15, seg = tid >> 4;
    float* p = smem + row * NSP + seg * 256;
    const float mx = rowmax[row];
    float sum = 0.0f;
    for (int j = 0; j < 256; ++j) {
      const float e = __expf(p[j] - mx);
      p[j] = e;
      sum += e;
    }
    red[row * 16 + seg] = sum;
  }
  __syncthreads();
  if (tid < 16) {
    float sum = 0.0f;
    for (int sg = 0; sg < 16; ++sg) sum += red[tid * 16 + sg];
    rowinv[tid] = 1.0f / sum;
  }
  __syncthreads();

  // column sums of the normalized block -> s
  for (int n = tid; n < NSP; n += 256) {
    float acc = 0.0f;
#pragma unroll
    for (int r = 0; r < 16; ++r) acc += smem[r * NSP + n] * rowinv[r];
    atomicAdd(&s[(size_t)b * NSP + n], acc);
  }
}

// ---------------------------------------------------------------------------
// dot_out[b,row] = X[b,row,:] . s[b,:]   (optionally mean_out = row mean)
// One wave per row, float4 (global_load_b128) streaming + prefetch; this
// kernel carries the dominant 537 MB pc2d read.
// ---------------------------------------------------------------------------
__global__ __launch_bounds__(256)
void rowdot_kernel(const float* __restrict__ X, const float* __restrict__ s,
                   float* __restrict__ dot_out, float* __restrict__ mean_out,
                   int C)
{
  const int lane = threadIdx.x & 31, wave = threadIdx.x >> 5;
  const int row = blockIdx.x * 8 + wave;
  const int b = blockIdx.y;
  const float4* __restrict__ xp =
      (const float4*)(X + (size_t)b * C * NSP + (size_t)row * NSP);
  const float4* __restrict__ sp = (const float4*)(s + (size_t)b * NSP);
  float acc = 0.0f, macc = 0.0f;
  for (int n = lane; n < NSP / 4; n += 32) {
    if (n + 64 < NSP / 4) __builtin_prefetch(xp + n + 64, 0, 0);
    const float4 v = xp[n];
    const float4 sv = sp[n];
    acc += v.x * sv.x + v.y * sv.y + v.z * sv.z + v.w * sv.w;
    macc += v.x + v.y + v.z + v.w;
  }
  acc = wave_reduce_sum(acc);
  macc = wave_reduce_sum(macc);
  if (lane == 0) {
    dot_out[(size_t)b * C + row] = acc;
    if (mean_out) mean_out[(size_t)b * C + row] = macc * (1.0f / NSP);
  }
}

// ---------------------------------------------------------------------------
// fused[b, 0:256]    = gamma*((Wvi@ivec)/N + bvi) + img_mean
// fused[b, 256:2304] = (Wvp@pvec)/N + bvp
// (Uses sum_n s[n] == N to fold the biases.)
// ---------------------------------------------------------------------------
__global__ __launch_bounds__(256)
void fuse_kernel(const float* __restrict__ Wvi, const float* __restrict__ bvi,
                 const float* __restrict__ Wvp, const float* __restrict__ bvp,
                 const float* __restrict__ gamma,
                 const float* __restrict__ ivec, const float* __restrict__ imean,
                 const float* __restrict__ pvec, float* __restrict__ fused)
{
  const int lane = threadIdx.x & 31, wave = threadIdx.x >> 5;
  const int c = blockIdx.x * 8 + wave;
  const int b = blockIdx.y;
  float r;
  if (c < CQ_DIM) {
    const float* w = Wvi + (size_t)c * CQ_DIM;
    const float* v = ivec + (size_t)b * CQ_DIM;
    float acc = 0.0f;
    for (int j = lane; j < CQ_DIM; j += 32) acc += w[j] * v[j];
    acc = wave_reduce_sum(acc);
    r = gamma[0] * (acc * (1.0f / NSP) + bvi[c]) + imean[(size_t)b * CQ_DIM + c];
  } else {
    const int cc = c - CQ_DIM;
    const float* w = Wvp + (size_t)cc * CK_DIM;
    const float* v = pvec + (size_t)b * CK_DIM;
    float acc = 0.0f;
    for (int j = lane; j < CK_DIM; j += 32) acc += w[j] * v[j];
    acc = wave_reduce_sum(acc);
    r = acc * (1.0f / NSP) + bvp[cc];
  }
  if (lane == 0) fused[(size_t)b * FUSED_DIM + c] = r;
}

// ---------------------------------------------------------------------------
// h = relu(W1@fused + b1); logits = W2@h + b2; out = log_softmax(logits)
// One block per batch element; tiny.
// ---------------------------------------------------------------------------
__global__ __launch_bounds__(256)
void head_kernel(const float* __restrict__ W1, const float* __restrict__ b1,
                 const float* __restrict__ W2, const float* __restrict__ b2,
                 const float* __restrict__ fused, float* __restrict__ out)
{
  __shared__ float sf[FUSED_DIM];
  __shared__ float sh[HID];
  __shared__ float slog[NCLS];
  const int b = blockIdx.x, tid = threadIdx.x;
  for (int i = tid; i < FUSED_DIM; i += 256) sf[i] = fused[(size_t)b * FUSED_DIM + i];
  __syncthreads();
  for (int j = tid; j < HID; j += 256) {
    const float* w = W1 + (size_t)j * FUSED_DIM;
    float acc = b1[j];
    for (int i = 0; i < FUSED_DIM; ++i) acc += w[i] * sf[i];
    sh[j] = fmaxf(acc, 0.0f);
  }
  __syncthreads();
  if (tid < NCLS) {
    const float* w = W2 + (size_t)tid * HID;
    float acc = b2[tid];
    for (int i = 0; i < HID; ++i) acc += w[i] * sh[i];
    slog[tid] = acc;
  }
  __syncthreads();
  if (tid == 0) {
    float mx = -3.4e38f;
    for (int i = 0; i < NCLS; ++i) mx = fmaxf(mx, slog[i]);
    float sum = 0.0f;
    for (int i = 0; i < NCLS; ++i) sum += __expf(slog[i] - mx);
    const float lse = mx + logf(sum);
    for (int i = 0; i < NCLS; ++i) out[(size_t)b * NCLS + i] = slog[i] - lse;
  }
}

extern "C" void kernel_launch(void* const* d_in, const int* in_sizes, int n_in,
                              void* d_out, int out_size, void* d_ws, size_t ws_size,
                              hipStream_t stream)
{
  (void)in_sizes; (void)n_in; (void)out_size; (void)ws_size;
  const float* img   = (const float*)d_in[0];
  const float* pc2d  = (const float*)d_in[1];
  const float* Wq    = (const float*)d_in[2];
  const float* bq    = (const float*)d_in[3];
  const float* Wk    = (const float*)d_in[4];
  const float* bk    = (const float*)d_in[5];
  const float* Wvi   = (const float*)d_in[6];
  const float* bvi   = (const float*)d_in[7];
  const float* Wvp   = (const float*)d_in[8];
  const float* bvp   = (const float*)d_in[9];
  const float* gamma = (const float*)d_in[10];
  const float* W1    = (const float*)d_in[11];
  const float* b1    = (const float*)d_in[12];
  const float* W2    = (const float*)d_in[13];
  const float* b2    = (const float*)d_in[14];
  float* out = (float*)d_out;

  float* ws = (float*)d_ws;
  size_t off = 0;
  float* qbuf  = ws + off; off += (size_t)BB * NSP * CQ_DIM;  // [B,N,CQ]
  float* kbuf  = ws + off; off += (size_t)BB * CQ_DIM * NSP;  // [B,CQ,N]
  float* sbuf  = ws + off; off += (size_t)BB * NSP;           // att col sums
  float* ivec  = ws + off; off += (size_t)BB * CQ_DIM;        // img @ s
  float* imean = ws + off; off += (size_t)BB * CQ_DIM;        // mean(img)
  float* pvec  = ws + off; off += (size_t)BB * CK_DIM;        // pc2d @ s
  float* fusedb= ws + off; off += (size_t)BB * FUSED_DIM;

  zero_kernel<<<(BB * NSP + 255) / 256, 256, 0, stream>>>(sbuf, BB * NSP);

  // q projection, stored transposed -> [B, N, CQ]
  conv1x1_wmma_kernel<<<dim3(NSP / 64, CQ_DIM / 64, BB), 128, 0, stream>>>(
      Wq, img, bq, qbuf, CQ_DIM, CQ_DIM, 1);
  // k projection -> [B, CQ, N]
  conv1x1_wmma_kernel<<<dim3(NSP / 64, CQ_DIM / 64, BB), 128, 0, stream>>>(
      Wk, pc2d, bk, kbuf, CQ_DIM, CK_DIM, 0);

  // QK^T + softmax, reduced to column sums s (256 KB LDS per workgroup)
  attn_colsum_kernel<<<dim3(NSP / 16, BB), 256, 16 * NSP * sizeof(float),
                       stream>>>(qbuf, kbuf, sbuf);

  rowdot_kernel<<<dim3(CQ_DIM / 8, BB), 256, 0, stream>>>(img, sbuf, ivec, imean, CQ_DIM);
  rowdot_kernel<<<dim3(CK_DIM / 8, BB), 256, 0, stream>>>(pc2d, sbuf, pvec, nullptr, CK_DIM);

  fuse_kernel<<<dim3(FUSED_DIM / 8, BB), 256, 0, stream>>>(
      Wvi, bvi, Wvp, bvp, gamma, ivec, imean, pvec, fusedb);
  head_kernel<<<BB, 256, 0, stream>>>(W1, b1, W2, b2, fusedb, out);
}